// XSimGCL_Encoder_46952582479982
// MI455X (gfx1250) — compile-verified
//
#include <hip/hip_runtime.h>
#include <math.h>

#define USER_NUM 100000
#define ITEM_NUM 50000
#define NTOT     150000
#define DD       64
#define NNZ_CNT  2000000
#define EPSP     0.2f

typedef __attribute__((ext_vector_type(2))) float v2f;
typedef __attribute__((ext_vector_type(8))) float v8f;

// ---------------------------------------------------------------------------
// Zero a float buffer (float4 granularity).
// ---------------------------------------------------------------------------
__global__ void zero_kernel(float4* __restrict__ p, long long n4) {
  long long i = (long long)blockIdx.x * blockDim.x + threadIdx.x;
  if (i < n4) p[i] = make_float4(0.f, 0.f, 0.f, 0.f);
}

// ---------------------------------------------------------------------------
// Non-returning device-scope f32 atomic add. Forced to the CDNA5 hardware
// instruction (L2 atomic units) via inline asm so it can never be expanded
// into a CAS loop. Non-returning -> tracked with STOREcnt only; S_ENDPGM's
// implicit wait-idle covers completion, stream order covers visibility.
// ---------------------------------------------------------------------------
__device__ __forceinline__ void gatomic_add(float* p, float v) {
#if defined(__gfx1250__)
  asm volatile("global_atomic_add_f32 %0, %1, off scope:SCOPE_DEV"
               :
               : "v"(p), "v"(v)
               : "memory");
#else
  atomicAdd(p, v);
#endif
}

// ---------------------------------------------------------------------------
// COO SpMM: y[r,:] += v * x[c,:].  16 threads per nonzero, float4 per thread.
// x (38.4MB) and y (38.4MB) are L2-resident on MI455X (192MB L2).
// ---------------------------------------------------------------------------
__global__ void spmm_atomic_kernel(const int* __restrict__ rows,
                                   const int* __restrict__ cols,
                                   const float* __restrict__ vals,
                                   const float* __restrict__ x,
                                   float* __restrict__ y, int nnz) {
  long long t = (long long)blockIdx.x * blockDim.x + threadIdx.x;
  int nz = (int)(t >> 4);
  if (nz >= nnz) return;
  int d4 = ((int)t & 15) << 2;
  int r = rows[nz];
  int c = cols[nz];
  float v = vals[nz];
  const float4 xv = *reinterpret_cast<const float4*>(x + (long long)c * DD + d4);
  float* out = y + (long long)r * DD + d4;
  gatomic_add(out + 0, v * xv.x);
  gatomic_add(out + 1, v * xv.y);
  gatomic_add(out + 2, v * xv.z);
  gatomic_add(out + 3, v * xv.w);
}

// ---------------------------------------------------------------------------
// Row-norm via V_WMMA_F32_16X16X4_F32.
// One wave handles 16 rows of noise[N,64]. For each K-chunk of 4, the per-lane
// float2 load (row = lane&15, k = k0 + 2*(lane>>4) + {0,1}) matches BOTH the
// 16x4 A layout and the 4x16 B layout, so WMMA(A=X, B=X) = X * X^T (Gram).
// Accumulating 16 chunks covers K=64; the Gram diagonal = squared row norms.
// Diagonal ownership (16x16 f32 C layout): lanes 0-7 -> (M=lane, vgpr=lane),
// lanes 24-31 -> (M=lane-16, vgpr=lane-24).
// Stores inv_norm = EPS / max(||row||, 1e-12).
// ---------------------------------------------------------------------------
__global__ void noise_norm_wmma_kernel(const float* __restrict__ noise,
                                       float* __restrict__ inv_norm,
                                       int ntiles) {
  int wave = blockIdx.x * (blockDim.x >> 5) + (threadIdx.x >> 5);
  if (wave >= ntiles) return;  // wave-uniform: EXEC stays all-ones for WMMA
  int lane  = threadIdx.x & 31;
  int row   = lane & 15;
  int khalf = lane >> 4;
  const float* base = noise + ((long long)wave * 16 + row) * DD + (khalf << 1);

  v8f c = {};
#pragma unroll
  for (int k0 = 0; k0 < DD; k0 += 4) {
    v2f a = *reinterpret_cast<const v2f*>(base + k0);
    c = __builtin_amdgcn_wmma_f32_16x16x4_f32(
        /*neg_a=*/false, a, /*neg_b=*/false, a,
        /*c_mod=*/(short)0, c, /*reuse_a=*/false, /*reuse_b=*/false);
  }

  int m = -1, v = 0;
  if (lane < 8)        { m = lane;      v = lane;      }
  else if (lane >= 24) { m = lane - 16; v = lane - 24; }
  if (m >= 0) {
    float s = c[0];
#pragma unroll
    for (int i = 1; i < 8; ++i)
      if (v == i) s = c[i];
    float nrm = sqrtf(s);
    inv_norm[wave * 16 + m] = EPSP / fmaxf(nrm, 1e-12f);
  }
}

// ---------------------------------------------------------------------------
// e = a*h - b*m; e += sign(e) * noise * inv_norm[row]; acc (=|+=) e.
// ---------------------------------------------------------------------------
__device__ __forceinline__ float sgnf(float x) {
  return (x > 0.f) ? 1.f : ((x < 0.f) ? -1.f : 0.f);
}

__global__ void combine_kernel(const float* __restrict__ h,
                               const float* __restrict__ m,
                               const float* __restrict__ noise,
                               const float* __restrict__ inv_norm,
                               float a, float b,
                               float* __restrict__ e_out,
                               float* __restrict__ acc,
                               int first, long long n4) {
  long long i = (long long)blockIdx.x * blockDim.x + threadIdx.x;
  if (i >= n4) return;
  long long base = i << 2;
  int row = (int)(base >> 6);  // D == 64
  float invn = inv_norm[row];

  float4 hv = *reinterpret_cast<const float4*>(h + base);
  float4 mv = *reinterpret_cast<const float4*>(m + base);
  float4 nv = *reinterpret_cast<const float4*>(noise + base);

  float4 ev;
  ev.x = a * hv.x - b * mv.x;  ev.x += sgnf(ev.x) * nv.x * invn;
  ev.y = a * hv.y - b * mv.y;  ev.y += sgnf(ev.y) * nv.y * invn;
  ev.z = a * hv.z - b * mv.z;  ev.z += sgnf(ev.z) * nv.z * invn;
  ev.w = a * hv.w - b * mv.w;  ev.w += sgnf(ev.w) * nv.w * invn;

  *reinterpret_cast<float4*>(e_out + base) = ev;

  if (first) {
    *reinterpret_cast<float4*>(acc + base) = ev;
  } else {
    float4 av = *reinterpret_cast<const float4*>(acc + base);
    av.x += ev.x; av.y += ev.y; av.z += ev.z; av.w += ev.w;
    *reinterpret_cast<float4*>(acc + base) = av;
  }
}

__global__ void scale_kernel(float4* __restrict__ p, float s, long long n4) {
  long long i = (long long)blockIdx.x * blockDim.x + threadIdx.x;
  if (i < n4) {
    float4 v = p[i];
    v.x *= s; v.y *= s; v.z *= s; v.w *= s;
    p[i] = v;
  }
}

// ---------------------------------------------------------------------------
extern "C" void kernel_launch(void* const* d_in, const int* in_sizes, int n_in,
                              void* d_out, int out_size, void* d_ws, size_t ws_size,
                              hipStream_t stream) {
  const float* user_emb = (const float*)d_in[0];
  const float* item_emb = (const float*)d_in[1];
  const int*   adj_r = (const int*)d_in[2];
  const int*   adj_c = (const int*)d_in[3];
  const float* adj_v = (const float*)d_in[4];
  const int*   ai_r  = (const int*)d_in[5];
  const int*   ai_c  = (const int*)d_in[6];
  const float* ai_v  = (const float*)d_in[7];
  const int*   aj_r  = (const int*)d_in[8];
  const int*   aj_c  = (const int*)d_in[9];
  const float* aj_v  = (const float*)d_in[10];
  const float* noises[4] = {(const float*)d_in[11], (const float*)d_in[12],
                            (const float*)d_in[13], (const float*)d_in[14]};

  const long long ND = (long long)NTOT * DD;        // 9.6M floats
  float* E     = (float*)d_ws;                      // e / m1 (liveness-reused)
  float* H     = E + ND;                            // h
  float* M     = H + ND;                            // m2
  float* NORMS = M + ND;                            // 4 * NTOT inv-norms

  dim3 blk(256);
  const long long n4 = ND / 4;                            // 2.4M float4
  const int gz = (int)((n4 + 255) / 256);                 // 9375 blocks
  const int gs = (int)(((long long)NNZ_CNT * 16 + 255) / 256); // 125000 blocks
  const int ntiles = NTOT / 16;                           // 9375 (exact)
  const int gn = (ntiles + 7) / 8;                        // 8 waves / block

  // e0 = concat(user_emb, item_emb)
  hipMemcpyAsync(E, user_emb, (size_t)USER_NUM * DD * sizeof(float),
                 hipMemcpyDeviceToDevice, stream);
  hipMemcpyAsync(E + (long long)USER_NUM * DD, item_emb,
                 (size_t)ITEM_NUM * DD * sizeof(float),
                 hipMemcpyDeviceToDevice, stream);

  // Per-noise inverse row norms (WMMA Gram diagonal)
  for (int k = 0; k < 4; ++k)
    noise_norm_wmma_kernel<<<gn, blk, 0, stream>>>(
        noises[k], NORMS + (long long)k * NTOT, ntiles);

  const float Av[4] = {1.1f, 2.1f, 1.1f, 1.05f};
  const float Bv[4] = {0.1f, 1.1f, 0.1f, 0.05f};

  for (int k = 0; k < 4; ++k) {
    // h = adj @ e        (E dead afterwards)
    zero_kernel<<<gz, blk, 0, stream>>>((float4*)H, n4);
    spmm_atomic_kernel<<<gs, blk, 0, stream>>>(adj_r, adj_c, adj_v, E, H, NNZ_CNT);
    // m1 = aj @ h   -> reuse E
    zero_kernel<<<gz, blk, 0, stream>>>((float4*)E, n4);
    spmm_atomic_kernel<<<gs, blk, 0, stream>>>(aj_r, aj_c, aj_v, H, E, NNZ_CNT);
    // m2 = ai @ m1
    zero_kernel<<<gz, blk, 0, stream>>>((float4*)M, n4);
    spmm_atomic_kernel<<<gs, blk, 0, stream>>>(ai_r, ai_c, ai_v, E, M, NNZ_CNT);
    // e_{k+1} = a*h - b*m + sign(.)*normalize(noise)*EPS ; acc += e_{k+1}
    combine_kernel<<<gz, blk, 0, stream>>>(H, M, noises[k],
                                           NORMS + (long long)k * NTOT,
                                           Av[k], Bv[k], E, (float*)d_out,
                                           (k == 0) ? 1 : 0, n4);
  }

  // final = mean of e1..e4
  scale_kernel<<<gz, blk, 0, stream>>>((float4*)d_out, 0.25f, n4);
}